// GCN_28741921145256
// MI455X (gfx1250) — compile-verified
//
#include <hip/hip_runtime.h>

typedef __attribute__((ext_vector_type(2))) float v2f;
typedef __attribute__((ext_vector_type(8))) float v8f;

#define N_NODES 100000
#define N_EDGES 3200000
#define G_GRAPHS 256
#define D_FEAT 128
#define EGO_DIM 64
#define MLP_H 128

// ---------------------------------------------------------------- helpers

__device__ inline void atomAddF(float* a, float v) {
  // lowers to global_atomic_add_f32 (no CAS loop) on gfx1250
  unsafeAtomicAdd(a, v);
}

__device__ inline void atomMaxF(float* addr, float v) {
  // classic float-max-via-int-atomics trick; storage initialized to -inf bits
  if (v >= 0.0f) atomicMax((int*)addr, __float_as_int(v));
  else           atomicMin((unsigned int*)addr, __float_as_uint(v));
}

// ---------------------------------------------------------------- degree / norm

__global__ __launch_bounds__(256) void deg_init_kernel(float* deg, int n) {
  int i = blockIdx.x * blockDim.x + threadIdx.x;
  if (i < n) deg[i] = 1.0f;                 // self loop counts once
}

__global__ __launch_bounds__(256) void deg_accum_kernel(const int* __restrict__ dst,
                                                        float* __restrict__ deg, int e) {
  int i = blockIdx.x * blockDim.x + threadIdx.x;
  if (i < e) atomAddF(&deg[dst[i]], 1.0f);
}

__global__ __launch_bounds__(256) void dinv_kernel(float* deg, int n) {
  int i = blockIdx.x * blockDim.x + threadIdx.x;
  if (i < n) deg[i] = rsqrtf(deg[i]);       // deg >= 1 always (self loops)
}

// ---------------------------------------------------------------- WMMA fp32 GEMM
// C[m, coff+n] = sum_k A[m*lda+k] * B[k*NCOLS+n] (+bias[n]) (+relu)
// One wave computes a 16-row strip across all NT 16-col tiles.
// B (K x NCOLS, K<=128, NCOLS<=128) is staged into LDS by the whole block.

template <int NT, bool BIAS, bool RELU>
__global__ __launch_bounds__(256)
void wmma_gemm_kernel(const float* __restrict__ A, int lda,
                      const float* __restrict__ B, int kdim,
                      const float* __restrict__ bias,
                      float* __restrict__ C, int ldc, int coff,
                      int mtiles) {
  constexpr int NCOLS = NT * 16;
  __shared__ float Bs[128 * 128];           // 64 KB of the 320 KB WGP LDS

  // cooperative stage of B into LDS (float4)
  {
    const float4* B4  = (const float4*)B;
    float4*       Bs4 = (float4*)Bs;
    int total4 = (kdim * NCOLS) >> 2;
    for (int i = threadIdx.x; i < total4; i += blockDim.x) Bs4[i] = B4[i];
  }
  __syncthreads();

  const int wave  = threadIdx.x >> 5;
  const int lane  = threadIdx.x & 31;
  const int mtile = blockIdx.x * (blockDim.x >> 5) + wave;
  if (mtile >= mtiles) return;              // wave-uniform: EXEC all-1s inside WMMA

  const int half = lane >> 4;               // 0: lanes 0-15, 1: lanes 16-31
  const int lrow = lane & 15;

  v8f acc[NT];
  const v8f zero = {0.f, 0.f, 0.f, 0.f, 0.f, 0.f, 0.f, 0.f};
#pragma unroll
  for (int t = 0; t < NT; ++t) acc[t] = zero;

  // A: 16x4 fp32 tile; lane holds row (lane&15), Ks half*2 .. half*2+1
  const float* arow = A + (size_t)(mtile * 16 + lrow) * lda + half * 2;

  for (int k0 = 0; k0 < kdim; k0 += 4) {
    float2 av = *(const float2*)(arow + k0);
    v2f a; a[0] = av.x; a[1] = av.y;
    const int kb = k0 + half * 2;
#pragma unroll
    for (int t = 0; t < NT; ++t) {
      v2f b;
      b[0] = Bs[(kb)     * NCOLS + t * 16 + lrow];
      b[1] = Bs[(kb + 1) * NCOLS + t * 16 + lrow];
      acc[t] = __builtin_amdgcn_wmma_f32_16x16x4_f32(
          /*neg_a=*/false, a, /*neg_b=*/false, b,
          /*c_mod=*/(short)0, acc[t], /*reuse_a=*/false, /*reuse_b=*/false);
    }
  }

  // C 16x16 fp32 layout: VGPR r -> M = half*8 + r, N = lrow
#pragma unroll
  for (int t = 0; t < NT; ++t) {
#pragma unroll
    for (int r = 0; r < 8; ++r) {
      int m = mtile * 16 + half * 8 + r;
      int n = t * 16 + lrow;
      float v = acc[t][r];
      if (BIAS) v += bias[n];
      if (RELU) v = fmaxf(v, 0.0f);
      C[(size_t)m * ldc + coff + n] = v;
    }
  }
}

// ---------------------------------------------------------------- layer pieces

// out = h_in + bias + dinv[i]^2 * tmp   (residual + bias + self-loop message)
__global__ __launch_bounds__(256)
void layer_init_kernel(const float* __restrict__ hin, const float* __restrict__ tmp,
                       const float* __restrict__ dinv, const float* __restrict__ bias,
                       float* __restrict__ out) {
  int idx = blockIdx.x * blockDim.x + threadIdx.x;   // over N * D/4
  if (idx >= N_NODES * (D_FEAT / 4)) return;
  int node = idx >> 5;                               // D/4 == 32
  int f4   = idx & 31;
  float di = dinv[node];
  float sl = di * di;
  float4 h = ((const float4*)hin)[idx];
  float4 t = ((const float4*)tmp)[idx];
  float4 b = ((const float4*)bias)[f4];
  float4 o;
  o.x = h.x + b.x + sl * t.x;
  o.y = h.y + b.y + sl * t.y;
  o.z = h.z + b.z + sl * t.z;
  o.w = h.w + b.w + sl * t.w;
  ((float4*)out)[idx] = o;
}

// one wave per edge; lane covers 4 features -> 4 float atomics
__global__ __launch_bounds__(256)
void edge_scatter_kernel(const float* __restrict__ tmp, const float* __restrict__ dinv,
                         const int* __restrict__ src, const int* __restrict__ dst,
                         float* __restrict__ out, int e) {
  int lane = threadIdx.x & 31;
  int eidx = blockIdx.x * (blockDim.x >> 5) + (threadIdx.x >> 5);
  if (eidx >= e) return;
  int s = src[eidx];
  int d = dst[eidx];
  float norm = dinv[s] * dinv[d];
  float4 v = ((const float4*)(tmp + (size_t)s * D_FEAT))[lane];
  float* op = out + (size_t)d * D_FEAT + lane * 4;
  atomAddF(op + 0, v.x * norm);
  atomAddF(op + 1, v.y * norm);
  atomAddF(op + 2, v.z * norm);
  atomAddF(op + 3, v.w * norm);
}

__global__ __launch_bounds__(256) void relu_kernel(float* h, int n4) {
  int i = blockIdx.x * blockDim.x + threadIdx.x;
  if (i >= n4) return;
  float4 v = ((float4*)h)[i];
  v.x = fmaxf(v.x, 0.f); v.y = fmaxf(v.y, 0.f);
  v.z = fmaxf(v.z, 0.f); v.w = fmaxf(v.w, 0.f);
  ((float4*)h)[i] = v;
}

// ---------------------------------------------------------------- pooling

__global__ __launch_bounds__(256)
void pool_init_kernel(float* sums, float* maxv, float* counts) {
  int i = blockIdx.x * blockDim.x + threadIdx.x;
  if (i < G_GRAPHS * D_FEAT) {
    sums[i] = 0.0f;
    ((unsigned int*)maxv)[i] = 0xFF800000u;          // -inf
  }
  if (i < G_GRAPHS) counts[i] = 0.0f;
}

__global__ __launch_bounds__(256)
void count_kernel(const int* __restrict__ batch, float* __restrict__ counts) {
  int i = blockIdx.x * blockDim.x + threadIdx.x;
  if (i < N_NODES) atomAddF(&counts[batch[i]], 1.0f);
}

__global__ __launch_bounds__(256)
void pool_accum_kernel(const float* __restrict__ h, const int* __restrict__ batch,
                       float* __restrict__ sums, float* __restrict__ maxv) {
  int idx = blockIdx.x * blockDim.x + threadIdx.x;   // over N * 32
  if (idx >= N_NODES * (D_FEAT / 4)) return;
  int node = idx >> 5;
  int f4   = idx & 31;
  int g = batch[node];
  float4 v = ((const float4*)h)[idx];
  float* sp = sums + (size_t)g * D_FEAT + f4 * 4;
  float* mp = maxv + (size_t)g * D_FEAT + f4 * 4;
  atomAddF(sp + 0, v.x); atomAddF(sp + 1, v.y);
  atomAddF(sp + 2, v.z); atomAddF(sp + 3, v.w);
  atomMaxF(mp + 0, v.x); atomMaxF(mp + 1, v.y);
  atomMaxF(mp + 2, v.z); atomMaxF(mp + 3, v.w);
}

__global__ __launch_bounds__(256)
void pool_final_kernel(const float* __restrict__ sums, const float* __restrict__ maxv,
                       const float* __restrict__ counts, float* __restrict__ out) {
  int idx = blockIdx.x * blockDim.x + threadIdx.x;   // over G * D
  if (idx >= G_GRAPHS * D_FEAT) return;
  int g = idx >> 7;
  int f = idx & 127;
  float c = fmaxf(counts[g], 1.0f);
  out[(size_t)g * (D_FEAT + EGO_DIM) + f] = 0.5f * (sums[idx] / c + maxv[idx]);
}

// ---------------------------------------------------------------- launch

extern "C" void kernel_launch(void* const* d_in, const int* in_sizes, int n_in,
                              void* d_out, int out_size, void* d_ws, size_t ws_size,
                              hipStream_t stream) {
  const float* x    = (const float*)d_in[0];
  const int*   ei   = (const int*)d_in[1];
  const int*   batch= (const int*)d_in[2];
  const float* ego  = (const float*)d_in[3];
  const float* W1   = (const float*)d_in[4];
  const float* b1   = (const float*)d_in[5];
  const float* W2   = (const float*)d_in[6];
  const float* b2   = (const float*)d_in[7];
  const float* W3   = (const float*)d_in[8];
  const float* b3   = (const float*)d_in[9];
  const float* mw1  = (const float*)d_in[10];
  const float* mb1  = (const float*)d_in[11];
  const float* mw2  = (const float*)d_in[12];
  const float* mb2  = (const float*)d_in[13];
  float* out = (float*)d_out;

  const int* src = ei;             // edge_index[0]
  const int* dst = ei + N_EDGES;   // edge_index[1]

  float* ws    = (float*)d_ws;
  float* hA    = ws;                               // N*D
  float* hB    = hA + (size_t)N_NODES * D_FEAT;    // N*D
  float* tmp   = hB + (size_t)N_NODES * D_FEAT;    // N*D
  float* dinv  = tmp + (size_t)N_NODES * D_FEAT;   // N
  float* sums  = dinv + N_NODES;                   // G*D
  float* maxv  = sums + G_GRAPHS * D_FEAT;         // G*D
  float* cnts  = maxv + G_GRAPHS * D_FEAT;         // G
  float* ehid  = cnts + G_GRAPHS;                  // G*MLP_H

  const int ND4     = N_NODES * (D_FEAT / 4);      // 3.2M
  const int MTILES  = N_NODES / 16;                // 6250 (exact)
  const int GB      = (MTILES + 7) / 8;            // 8 waves / block

  // degree -> dinv
  deg_init_kernel<<<(N_NODES + 255) / 256, 256, 0, stream>>>(dinv, N_NODES);
  deg_accum_kernel<<<(N_EDGES + 255) / 256, 256, 0, stream>>>(dst, dinv, N_EDGES);
  dinv_kernel<<<(N_NODES + 255) / 256, 256, 0, stream>>>(dinv, N_NODES);

  // --- layer 1: x -> hA
  wmma_gemm_kernel<8, false, false><<<GB, 256, 0, stream>>>(x, D_FEAT, W1, D_FEAT,
                                                            nullptr, tmp, D_FEAT, 0, MTILES);
  layer_init_kernel<<<(ND4 + 255) / 256, 256, 0, stream>>>(x, tmp, dinv, b1, hA);
  edge_scatter_kernel<<<(N_EDGES + 7) / 8, 256, 0, stream>>>(tmp, dinv, src, dst, hA, N_EDGES);
  relu_kernel<<<(ND4 + 255) / 256, 256, 0, stream>>>(hA, ND4);

  // --- layer 2: hA -> hB
  wmma_gemm_kernel<8, false, false><<<GB, 256, 0, stream>>>(hA, D_FEAT, W2, D_FEAT,
                                                            nullptr, tmp, D_FEAT, 0, MTILES);
  layer_init_kernel<<<(ND4 + 255) / 256, 256, 0, stream>>>(hA, tmp, dinv, b2, hB);
  edge_scatter_kernel<<<(N_EDGES + 7) / 8, 256, 0, stream>>>(tmp, dinv, src, dst, hB, N_EDGES);
  relu_kernel<<<(ND4 + 255) / 256, 256, 0, stream>>>(hB, ND4);

  // --- layer 3 (no relu): hB -> hA
  wmma_gemm_kernel<8, false, false><<<GB, 256, 0, stream>>>(hB, D_FEAT, W3, D_FEAT,
                                                            nullptr, tmp, D_FEAT, 0, MTILES);
  layer_init_kernel<<<(ND4 + 255) / 256, 256, 0, stream>>>(hB, tmp, dinv, b3, hA);
  edge_scatter_kernel<<<(N_EDGES + 7) / 8, 256, 0, stream>>>(tmp, dinv, src, dst, hA, N_EDGES);

  // --- pooling (mean + max) / 2 -> out[:, 0:128]
  pool_init_kernel<<<(G_GRAPHS * D_FEAT + 255) / 256, 256, 0, stream>>>(sums, maxv, cnts);
  count_kernel<<<(N_NODES + 255) / 256, 256, 0, stream>>>(batch, cnts);
  pool_accum_kernel<<<(ND4 + 255) / 256, 256, 0, stream>>>(hA, batch, sums, maxv);
  pool_final_kernel<<<(G_GRAPHS * D_FEAT + 255) / 256, 256, 0, stream>>>(sums, maxv, cnts, out);

  // --- ego MLP: relu(ego @ mw1 + mb1) @ mw2 + mb2 -> out[:, 128:192]
  // 256 rows -> 16 m-tiles -> 2 blocks of 8 waves
  wmma_gemm_kernel<8, true, true><<<2, 256, 0, stream>>>(ego, EGO_DIM, mw1, EGO_DIM,
                                                         mb1, ehid, MLP_H, 0, G_GRAPHS / 16);
  wmma_gemm_kernel<4, true, false><<<2, 256, 0, stream>>>(ehid, MLP_H, mw2, MLP_H,
                                                          mb2, out, D_FEAT + EGO_DIM, D_FEAT,
                                                          G_GRAPHS / 16);
}